// BayesianCategoricalCrossEntropy_KendallAndGalsVersion_5334349381991
// MI455X (gfx1250) — compile-verified
//
#include <hip/hip_runtime.h>
#include <stdint.h>

#define NCLS 1000
#define NBATCH 2048
#define NT 100

typedef __attribute__((ext_vector_type(2))) float v2f;
typedef __attribute__((ext_vector_type(8))) float v8f;

// ---------------------------------------------------------------------------
// Phase 1: per-sample stats. One wave handles a 16-row tile.
//   - scan pass: row max, label (argmax of one-hot), x0 = pred[lab]
//   - WMMA pass: row sums of exp(pred - M) via V_WMMA_F32_16X16X4_F32 x ones,
//     dual accumulators to break the WMMA->WMMA serial dependency chain.
// ---------------------------------------------------------------------------
__global__ __launch_bounds__(256) void bcce_phase1(
    const float* __restrict__ tru,        // [B, C] one-hot
    const float* __restrict__ pv,         // [B, 2C] = [pred | var]
    float4* __restrict__ stats,           // [B] {M, S_rest, x0, sigma}
    int* __restrict__ labels)             // [B]
{
    __shared__ float lds_sum[8][16];
    const int lane = threadIdx.x & 31;
    const int wave = threadIdx.x >> 5;
    const int rowBase = blockIdx.x * 128 + wave * 16;
    const int r = lane & 15;       // row within tile (matrix M index)
    const int h = lane >> 4;       // lane-half selects K columns / row half
    const int row = rowBase + r;

    // ---- scan pass: lane handles columns [h*500, h*500+500) of its row ----
    const float* tp = tru + (size_t)row * NCLS + h * 500;
    const float* pp = pv  + (size_t)row * (2 * NCLS) + h * 500;
    float m = -3.4e38f;
    int   lab = -1;
    float x0  = 0.0f;
    for (int j = 0; j < 500; j += 4) {
        __builtin_prefetch(pp + j + 128, 0, 0);   // global_prefetch_b8
        __builtin_prefetch(tp + j + 128, 0, 0);
        const float4 p = *(const float4*)(pp + j);
        const float4 t = *(const float4*)(tp + j);
        m = fmaxf(fmaxf(fmaxf(m, p.x), p.y), fmaxf(p.z, p.w));
        if (t.x > 0.5f) { lab = h * 500 + j + 0; x0 = p.x; }
        if (t.y > 0.5f) { lab = h * 500 + j + 1; x0 = p.y; }
        if (t.z > 0.5f) { lab = h * 500 + j + 2; x0 = p.z; }
        if (t.w > 0.5f) { lab = h * 500 + j + 3; x0 = p.w; }
    }
    // combine the two 500-column halves (lane L <-> lane L^16)
    m = fmaxf(m, __shfl_xor(m, 16));
    const int   lab2 = __shfl_xor(lab, 16);
    const float x02  = __shfl_xor(x0, 16);
    if (lab < 0) { lab = lab2; x0 = x02; }

    // ---- WMMA pass: D += exp(A) x ones ; two independent accumulators ----
    // A layout (ISA 7.12.2): lane L<16 holds (M=L, K=0/1) in v0/v1,
    //                        lane L>=16 holds (M=L-16, K=2/3).
    v2f ones2; ones2.x = 1.0f; ones2.y = 1.0f;
    v8f acc0 = {};
    v8f acc1 = {};
    const float* prow = pv + (size_t)row * (2 * NCLS) + 2 * h;
    for (int k = 0; k < NCLS; k += 8) {          // 1000 = 8 * 125, exact
        const float2 p0 = *(const float2*)(prow + k);
        const float2 p1 = *(const float2*)(prow + k + 4);
        v2f a0, a1;
        a0.x = __expf(p0.x - m);
        a0.y = __expf(p0.y - m);
        a1.x = __expf(p1.x - m);
        a1.y = __expf(p1.y - m);
        acc0 = __builtin_amdgcn_wmma_f32_16x16x4_f32(
            false, a0, false, ones2, (short)0, acc0, false, false);
        acc1 = __builtin_amdgcn_wmma_f32_16x16x4_f32(
            false, a1, false, ones2, (short)0, acc1, false, false);
    }
    #pragma unroll
    for (int v = 0; v < 8; v++) acc0[v] += acc1[v];

    // D layout: lanes 0-15 vgpr v -> row v ; lanes 16-31 vgpr v -> row v+8.
    // All 16 columns identical (ones B), so any lane of the half works.
    if (lane == 0 || lane == 16) {
        const int base = (lane >> 4) * 8;
        #pragma unroll
        for (int v = 0; v < 8; v++) lds_sum[wave][base + v] = acc0[v];
    }
    __syncthreads();

    if (lane < 16) {
        const float S_all  = lds_sum[wave][lane];
        const float S_rest = S_all - __expf(x0 - m);
        const float sigma  = sqrtf(pv[(size_t)row * (2 * NCLS) + NCLS + lab]);
        stats[row]  = make_float4(m, S_rest, x0, sigma);
        labels[row] = lab;
    }
}

// ---------------------------------------------------------------------------
// Phase 2: per-sample MC loop. JAX threefry2x32 (key(42) -> {0,42}) + erfinv.
// ---------------------------------------------------------------------------
__device__ __forceinline__ uint32_t rotl32(uint32_t x, int n) {
    return (x << n) | (x >> (32 - n));
}

__device__ __forceinline__ void threefry2x32_42(uint32_t c0, uint32_t c1,
                                                uint32_t& o0, uint32_t& o1) {
    const uint32_t ks0 = 0u, ks1 = 42u, ks2 = 0x1BD11BDAu ^ 0u ^ 42u;
    uint32_t x0 = c0 + ks0, x1 = c1 + ks1;
#define TF_R4(ra, rb, rc, rd)                              \
    x0 += x1; x1 = rotl32(x1, ra); x1 ^= x0;               \
    x0 += x1; x1 = rotl32(x1, rb); x1 ^= x0;               \
    x0 += x1; x1 = rotl32(x1, rc); x1 ^= x0;               \
    x0 += x1; x1 = rotl32(x1, rd); x1 ^= x0;
    TF_R4(13, 15, 26, 6);  x0 += ks1; x1 += ks2 + 1u;
    TF_R4(17, 29, 16, 24); x0 += ks2; x1 += ks0 + 2u;
    TF_R4(13, 15, 26, 6);  x0 += ks0; x1 += ks1 + 3u;
    TF_R4(17, 29, 16, 24); x0 += ks1; x1 += ks2 + 4u;
    TF_R4(13, 15, 26, 6);  x0 += ks2; x1 += ks0 + 5u;
#undef TF_R4
    o0 = x0; o1 = x1;
}

__device__ __forceinline__ float erfinv_f(float x) {
    // Giles (2012), single-precision approximation.
    float w = -__logf((1.0f - x) * (1.0f + x));
    float p;
    if (w < 5.0f) {
        w -= 2.5f;
        p = 2.81022636e-08f;
        p = fmaf(p, w, 3.43273939e-07f);
        p = fmaf(p, w, -3.5233877e-06f);
        p = fmaf(p, w, -4.39150654e-06f);
        p = fmaf(p, w, 0.00021858087f);
        p = fmaf(p, w, -0.00125372503f);
        p = fmaf(p, w, -0.00417768164f);
        p = fmaf(p, w, 0.246640727f);
        p = fmaf(p, w, 1.50140941f);
    } else {
        w = sqrtf(w) - 3.0f;
        p = -0.000200214257f;
        p = fmaf(p, w, 0.000100950558f);
        p = fmaf(p, w, 0.00134934322f);
        p = fmaf(p, w, -0.00367342844f);
        p = fmaf(p, w, 0.00573950773f);
        p = fmaf(p, w, -0.0076224613f);
        p = fmaf(p, w, 0.00943887047f);
        p = fmaf(p, w, 1.00167406f);
        p = fmaf(p, w, 2.83297682f);
    }
    return p * x;
}

__global__ __launch_bounds__(256) void bcce_phase2(
    const float4* __restrict__ stats, const int* __restrict__ labels,
    float* __restrict__ means)
{
    const int b = blockIdx.x * blockDim.x + threadIdx.x;
    if (b >= NBATCH) return;
    const float4 s = stats[b];
    const int lab = labels[b];
    const float M = s.x, Srest = s.y, x0 = s.z, sigma = s.w;
    const uint64_t half = (uint64_t)NT * NBATCH * NCLS / 2;  // 102,400,000
    const float lo = -0.99999994f;                            // nextafter(-1,0)
    float accsum = 0.0f;
    for (int t = 0; t < NT; t++) {
        const uint64_t i = ((uint64_t)t * NBATCH + b) * (uint64_t)NCLS + lab;
        const bool first = (i < half);
        const uint32_t c0 = first ? (uint32_t)i : (uint32_t)(i - half);
        const uint32_t c1 = first ? (uint32_t)(i + half) : (uint32_t)i;
        uint32_t o0, o1;
        threefry2x32_42(c0, c1, o0, o1);
        const uint32_t bits = first ? o0 : o1;
        const float f01 = __uint_as_float((bits >> 9) | 0x3f800000u) - 1.0f; // [0,1)
        const float u = f01 * (1.0f - lo) + lo;                              // [lo,1)
        const float n = 1.4142135623730951f * erfinv_f(u);
        const float x = x0 + n * sigma;
        accsum += M + __logf(Srest + __expf(x - M)) - x;
    }
    means[b] = accsum * (1.0f / NT);
}

// ---------------------------------------------------------------------------
// Phase 3: deterministic tree reduction of the 2048 per-sample means.
// ---------------------------------------------------------------------------
__global__ __launch_bounds__(256) void bcce_phase3(
    const float* __restrict__ means, float* __restrict__ out)
{
    __shared__ float red[256];
    float s = 0.0f;
    for (int i = threadIdx.x; i < NBATCH; i += 256) s += means[i];
    red[threadIdx.x] = s;
    __syncthreads();
    for (int k = 128; k > 0; k >>= 1) {
        if ((int)threadIdx.x < k) red[threadIdx.x] += red[threadIdx.x + k];
        __syncthreads();
    }
    if (threadIdx.x == 0) out[0] = red[0] * (1.0f / NBATCH);
}

// ---------------------------------------------------------------------------
extern "C" void kernel_launch(void* const* d_in, const int* in_sizes, int n_in,
                              void* d_out, int out_size, void* d_ws, size_t ws_size,
                              hipStream_t stream) {
    (void)in_sizes; (void)n_in; (void)out_size; (void)ws_size;
    const float* tru = (const float*)d_in[0];   // true: [2048,1000] f32
    const float* pv  = (const float*)d_in[1];   // pred_var: [2048,2000] f32

    float4* stats = (float4*)d_ws;                                     // 32 KB
    int* labels   = (int*)((char*)d_ws + NBATCH * sizeof(float4));     //  8 KB
    float* means  = (float*)((char*)d_ws + NBATCH * sizeof(float4)
                                         + NBATCH * sizeof(int));      //  8 KB

    bcce_phase1<<<NBATCH / 128, 256, 0, stream>>>(tru, pv, stats, labels);
    bcce_phase2<<<NBATCH / 256, 256, 0, stream>>>(stats, labels, means);
    bcce_phase3<<<1, 256, 0, stream>>>(means, (float*)d_out);
}